// SimpleGCNLayer_8306466750606
// MI455X (gfx1250) — compile-verified
//
#include <hip/hip_runtime.h>
#include <hip/hip_bf16.h>

#define NN    2048
#define DD    128
#define BB    8
#define KSTEP 64
#define EPSF  1e-6f

typedef __attribute__((ext_vector_type(2))) float v2f;
typedef __attribute__((ext_vector_type(8))) float v8f;

// ---------------------------------------------------------------------------
// Kernel 1: per-row masked degree + normalization scales.
//   s[b,n]  = m * deg^-1/2      (row/col scale)
//   dt[b,n] = m * deg^-1        (diagonal contribution d*m*d)
// One wave32 per row, 8 waves per 256-thread block.
// ---------------------------------------------------------------------------
__global__ __launch_bounds__(256) void gcn_deg_kernel(
    const float* __restrict__ A, const int* __restrict__ mask,
    float* __restrict__ sbuf, float* __restrict__ dtbuf)
{
    const int tid  = threadIdx.x;
    const int wave = tid >> 5;
    const int lane = tid & 31;
    const int row  = blockIdx.x * 8 + wave;      // flat over (b, n): 0..16383
    const int b    = row >> 11;
    const int n    = row & (NN - 1);

    const float* __restrict__ Ar = A + (size_t)row * NN;
    const int*   __restrict__ mb = mask + b * NN;

    float sum = 0.0f;
    for (int j = lane; j < NN; j += 32)
        sum += Ar[j] * (float)mb[j];

    #pragma unroll
    for (int off = 16; off > 0; off >>= 1)
        sum += __shfl_xor(sum, off, 32);

    const float mr  = (float)mb[n];
    float deg = mr * sum + mr;                   // masked row sum + self loop
    deg = deg < EPSF ? EPSF : deg;
    const float dinv = rsqrtf(deg);

    if (lane == 0) {
        sbuf [row] = mr * dinv;                  // s  = m * d
        dtbuf[row] = mr * dinv * dinv;           // dt = m * d^2
    }
}

// ---------------------------------------------------------------------------
// Kernel 2: H = A_norm @ X via V_WMMA_F32_16X16X4_F32, normalization hoisted:
//   H[r,:] = s_r * sum_c A[r,c] * (s_c * X[c,:])  +  dt_r * X[r,:]
// Block = 128 threads (4 waves); wave owns a 16x128 output tile.
// X chunk (64x128) is staged in LDS prescaled by s_c and K-PAIR INTERLEAVED
// so each lane's B operand is one aligned ds_load_b64 (no packing movs).
// Inner loop per K-step of 4: 1 global b64 (A == WMMA A-operand directly),
// 8 ds b64, 8 wmma. Row scale + diagonal applied in the epilogue.
// ---------------------------------------------------------------------------
__global__ __launch_bounds__(128) void gcn_spmm_kernel(
    const float* __restrict__ A, const float* __restrict__ X,
    const float* __restrict__ sbuf, const float* __restrict__ dtbuf,
    float* __restrict__ out)
{
    // layout: sX2[p*256 + n*2 + e], p = (k - k0)/2, e = k&1   (32 KB)
    __shared__ __align__(16) float sX2[(KSTEP / 2) * DD * 2];

    const int b    = blockIdx.y;
    const int tid  = threadIdx.x;
    const int wave = tid >> 5;
    const int lane = tid & 31;
    const int m    = lane & 15;          // M index for A / N index for B,C
    const int kb   = (lane >> 4) << 1;   // K base within 4-wide step: 0 or 2

    const int row = blockIdx.x * 64 + wave * 16 + m;

    const float* __restrict__ Arow  = A + ((size_t)b * NN + row) * NN;
    const float* __restrict__ Xb    = X + (size_t)b * NN * DD;
    const float* __restrict__ sb    = sbuf  + (size_t)b * NN;
    const float* __restrict__ dtb   = dtbuf + (size_t)b * NN;

    v8f acc[8];
    #pragma unroll
    for (int t = 0; t < 8; ++t) acc[t] = {};

    for (int k0 = 0; k0 < NN; k0 += KSTEP) {
        __syncthreads();
        // Stage X[b, k0:k0+64, :] prescaled by s_c, pair-interleaved.
        // 1024 units of (k-pair x 4 columns); 8 units per thread.
        #pragma unroll
        for (int i = 0; i < 8; ++i) {
            const int u  = tid + i * 128;
            const int p  = u >> 5;            // k-pair 0..31
            const int ng = u & 31;            // column group 0..31 (4 cols)
            const int k  = k0 + 2 * p;
            const float sa = sb[k];
            const float sc = sb[k + 1];
            const float4 ra = *(const float4*)(Xb + (size_t)k * DD + 4 * ng);
            const float4 rb = *(const float4*)(Xb + (size_t)(k + 1) * DD + 4 * ng);
            float4 w0, w1;
            w0.x = ra.x * sa; w0.y = rb.x * sc; w0.z = ra.y * sa; w0.w = rb.y * sc;
            w1.x = ra.z * sa; w1.y = rb.z * sc; w1.z = ra.w * sa; w1.w = rb.w * sc;
            float4* dst = (float4*)(sX2 + p * 256 + 8 * ng);
            dst[0] = w0;
            dst[1] = w1;
        }
        __syncthreads();

        // Prefetch next A chunk: 32 lanes cover 16 rows x 2 cachelines.
        if (k0 + KSTEP < NN)
            __builtin_prefetch(Arow + k0 + KSTEP + ((lane >> 4) << 5), 0, 1);

        for (int kk = 0; kk < KSTEP; kk += 4) {
            // A operand: raw adjacency, two contiguous f32 per lane.
            const v2f av = *(const v2f*)(Arow + k0 + kk + kb);
            const float* bbase = sX2 + ((kk >> 1) + (lane >> 4)) * 256 + m * 2;
            #pragma unroll
            for (int t = 0; t < 8; ++t) {
                const v2f bv = *(const v2f*)(bbase + t * 32);
                acc[t] = __builtin_amdgcn_wmma_f32_16x16x4_f32(
                    false, av, false, bv, (short)0, acc[t], false, false);
            }
        }
    }

    // Epilogue: out[rM,n] = s_rM * acc + dt_rM * X[rM,n]
    // C layout: VGPR v, lanes 0-15 -> M=v, lanes 16-31 -> M=v+8; N = lane&15.
    const int mbase = blockIdx.x * 64 + wave * 16 + ((lane >> 4) << 3);
    float* __restrict__ ob = out + (size_t)b * NN * DD;
    #pragma unroll
    for (int v = 0; v < 8; ++v) {
        const int   rM = mbase + v;
        const float sv = sb[rM];
        const float dv = dtb[rM];
        #pragma unroll
        for (int t = 0; t < 8; ++t) {
            const int n = t * 16 + m;
            const float xv = Xb[(size_t)rM * DD + n];
            ob[(size_t)rM * DD + n] = sv * acc[t][v] + dv * xv;
        }
    }
}

// ---------------------------------------------------------------------------
// Kernel 3: out = relu(H @ W^T + b), in-place on d_out.
// Wave owns 16 rows x 128 cols; K = 128 (32 WMMAs per N-tile).
// A and B operands are direct aligned b64 loads (no packing movs).
// ---------------------------------------------------------------------------
__global__ __launch_bounds__(128) void gcn_mlp_kernel(
    const float* __restrict__ W, const float* __restrict__ bias,
    float* __restrict__ out)
{
    const int tid  = threadIdx.x;
    const int wave = tid >> 5;
    const int lane = tid & 31;
    const int m    = lane & 15;
    const int kb   = (lane >> 4) << 1;

    const size_t rowbase = (size_t)blockIdx.x * 64 + (size_t)wave * 16;
    const float* __restrict__ Hrow = out + (rowbase + m) * DD;

    v8f acc[8];
    #pragma unroll
    for (int t = 0; t < 8; ++t) acc[t] = {};

    #pragma unroll 4
    for (int k0 = 0; k0 < DD; k0 += 4) {
        const v2f av = *(const v2f*)(Hrow + k0 + kb);
        #pragma unroll
        for (int t = 0; t < 8; ++t) {
            const v2f bv = *(const v2f*)(W + (size_t)(t * 16 + m) * DD + k0 + kb);
            acc[t] = __builtin_amdgcn_wmma_f32_16x16x4_f32(
                false, av, false, bv, (short)0, acc[t], false, false);
        }
    }

    const int srow = (lane >> 4) << 3;
    #pragma unroll
    for (int t = 0; t < 8; ++t) {
        const float bb = bias[t * 16 + m];
        #pragma unroll
        for (int v = 0; v < 8; ++v) {
            float val = acc[t][v] + bb;
            val = val > 0.0f ? val : 0.0f;
            out[(rowbase + srow + v) * DD + t * 16 + m] = val;
        }
    }
}

// ---------------------------------------------------------------------------
// Host-side launcher. Inputs: X(0), A(1), mask(2), W(3), b(4).
// ---------------------------------------------------------------------------
extern "C" void kernel_launch(void* const* d_in, const int* in_sizes, int n_in,
                              void* d_out, int out_size, void* d_ws, size_t ws_size,
                              hipStream_t stream) {
    (void)in_sizes; (void)n_in; (void)out_size; (void)ws_size;

    const float* X    = (const float*)d_in[0];
    const float* A    = (const float*)d_in[1];
    const int*   mask = (const int*)  d_in[2];
    const float* W    = (const float*)d_in[3];
    const float* bias = (const float*)d_in[4];
    float*       out  = (float*)d_out;

    float* sbuf  = (float*)d_ws;             // 16384 f32
    float* dtbuf = sbuf + (size_t)BB * NN;   // 16384 f32

    // 1) degree + scales: one wave per row, 8 rows per block.
    gcn_deg_kernel<<<(BB * NN) / 8, 256, 0, stream>>>(A, mask, sbuf, dtbuf);

    // 2) normalized SpMM via f32 WMMA: 64 rows per block, grid (N/64, B).
    gcn_spmm_kernel<<<dim3(NN / 64, BB), 128, 0, stream>>>(A, X, sbuf, dtbuf, out);

    // 3) dense head + bias + relu, in place: 64 rows per block.
    gcn_mlp_kernel<<<(BB * NN) / 64, 128, 0, stream>>>(W, bias, out);
}